// RoIPool_41918880808971
// MI455X (gfx1250) — compile-verified
//
#include <hip/hip_runtime.h>
#include <math.h>

// RoIPool forward, float32, NCHW.
// Shapes from reference: feat (4,256,50,50), rois (128,5), out (128,256,7,7).
// Design notes (MI455X / gfx1250):
//  - No matmul structure -> WMMA inapplicable; workload is a latency-bound
//    gather/max over an L2-resident (10 MB << 192 MB) feature map.
//  - C*P*P = 12544 = 49*256  ==> each 256-thread block maps to exactly one
//    RoI k, so the RoI descriptor decode is fully scalar (s_load + SALU).
//  - w-loop fully unrolled + h-loop unrolled x2, all per-lane predicated:
//    up to 2*KW = 20 INDEPENDENT exec-masked global_load_b32s issue
//    back-to-back (pipelined in LOADcnt) before any wait/reduce.
//  - global_prefetch_b8 (gfx1250) runs 2 rows ahead of the load batch.
//  - 8 wave32 waves/block for additional latency cover; stores are fully
//    coalesced (consecutive lanes -> consecutive output dwords).

namespace {
constexpr int   PP     = 7;        // pooled size
constexpr int   BINS   = PP * PP;  // 49
constexpr float RSCALE = 0.0625f;  // 1/16 spatial scale
constexpr int   KHMAX  = 10;
constexpr int   KWMAX  = 10;
}

__global__ __launch_bounds__(256) void roipool_fwd_kernel(
    const float* __restrict__ feat,   // [N,C,H,W]
    const float* __restrict__ rois,   // [K,5] = (batch, x1, y1, x2, y2)
    float* __restrict__ out,          // [K,C,7,7]
    int C, int H, int W)
{
    const int perK_blocks = (C * BINS) / 256;      // 49 blocks per RoI
    const int k   = blockIdx.x / perK_blocks;      // uniform across block
    const int off = (blockIdx.x % perK_blocks) * 256 + threadIdx.x; // [0, C*49)
    const int c   = off / BINS;
    const int pp  = off - c * BINS;
    const int ph  = pp / PP;
    const int pw  = pp - ph * PP;

    // RoI descriptor: k is SGPR-uniform -> scalar loads + scalar ALU.
    const float* r = rois + (size_t)k * 5;
    const int b   = (int)r[0];
    const int rsw = (int)rintf(r[1] * RSCALE);   // round-half-even == jnp.round
    const int rsh = (int)rintf(r[2] * RSCALE);
    const int rew = (int)rintf(r[3] * RSCALE);
    const int reh = (int)rintf(r[4] * RSCALE);
    const int roi_w = max(rew - rsw + 1, 1);
    const int roi_h = max(reh - rsh + 1, 1);
    const float bin_w = (float)roi_w / (float)PP;
    const float bin_h = (float)roi_h / (float)PP;

    // Window bounds, fp32 math to match JAX, clipped to [0, H] / [0, W],
    // then capped at KH/KW exactly like the reference's fixed-size gather.
    int hs = min(max((int)floorf((float)ph * bin_h) + rsh, 0), H);
    int he = min(max((int)ceilf(((float)ph + 1.0f) * bin_h) + rsh, 0), H);
    int ws = min(max((int)floorf((float)pw * bin_w) + rsw, 0), W);
    int we = min(max((int)ceilf(((float)pw + 1.0f) * bin_w) + rsw, 0), W);
    he = min(he, hs + KHMAX);
    we = min(we, ws + KWMAX);

    const int  wcount = we - ws;                  // 0..10
    const bool any    = (he > hs) && (wcount > 0);

    float m = -INFINITY;
    if (any) {
        const float* __restrict__ plane =
            feat + ((size_t)b * C + (size_t)c) * (size_t)(H * W);
        for (int h = hs; h < he; h += 2) {
            const float* __restrict__ row0 = plane + (size_t)h * W + ws;
            const bool has1 = (h + 1) < he;
            // gfx1250 global_prefetch_b8: pull rows h+2 / h+3 toward L0/L2,
            // two rows ahead of the in-flight load batch.
            if (h + 2 < he) __builtin_prefetch(row0 + 2 * W, 0, 3);
            if (h + 3 < he) __builtin_prefetch(row0 + 3 * W, 0, 3);

            // 2*KW independent, per-lane-predicated loads: all issue into
            // LOADcnt before the reduction waits. Every executed load is
            // provably in-bounds (h<he<=H, ws+j<we<=W).
            float v0[KWMAX], v1[KWMAX];
#pragma unroll
            for (int j = 0; j < KWMAX; ++j) {
                v0[j] = -INFINITY;
                if (j < wcount) v0[j] = row0[j];
            }
#pragma unroll
            for (int j = 0; j < KWMAX; ++j) {
                v1[j] = -INFINITY;
                if (has1 && j < wcount) v1[j] = row0[W + j];
            }
#pragma unroll
            for (int j = 0; j < KWMAX; ++j) {
                m = fmaxf(m, fmaxf(v0[j], v1[j]));
            }
        }
    }

    out[((size_t)k * C + (size_t)c) * BINS + pp] = any ? m : 0.0f;
}

extern "C" void kernel_launch(void* const* d_in, const int* in_sizes, int n_in,
                              void* d_out, int out_size, void* d_ws, size_t ws_size,
                              hipStream_t stream) {
    (void)n_in; (void)d_ws; (void)ws_size;
    const float* feat = (const float*)d_in[0];
    const float* rois = (const float*)d_in[1];
    float* out = (float*)d_out;

    const int K = in_sizes[1] / 5;                 // 128
    const int C = out_size / (K * BINS);           // 256
    const int H = 50, W = 50;                      // fixed by reference

    const int total  = K * C * BINS;               // 1,605,632
    const int blocks = total / 256;                // 6,272 (exact: 49 | C*BINS/256)
    roipool_fwd_kernel<<<blocks, 256, 0, stream>>>(feat, rois, out, C, H, W);
}